// GMMResBlock_12773232739126
// MI455X (gfx1250) — compile-verified
//
#include <hip/hip_runtime.h>

// GMMConv residual block for MI455X (gfx1250, wave32).
// Pipeline:
//   1) hipMemsetAsync: zero acc/cnt scratch
//   2) gmm_node_gemm_wmma: y = x @ g  (N x 192) and out = x @ root + bias + x
//      using V_WMMA_F32_16X16X4_F32 (full fp32 precision; compute is not the
//      bottleneck, so no reason to downcast)
//   3) gmm_edge_scatter: per-edge Gaussian weights + weighted gather from y
//      (L2-resident) + atomic scatter-add into acc, count into cnt
//   4) gmm_finalize: out = silu(acc/max(cnt,1) + out), in place

typedef __attribute__((ext_vector_type(2))) float v2f;
typedef __attribute__((ext_vector_type(8))) float v8f;

#define D_IN   64
#define YCOLS  192   // K * D
#define EPS_G  1e-14f

// One wave (32 threads) computes one 16x16 f32 tile of [y | xr].
// grid = (ceil(N/16), 16): col tiles 0..11 -> g (192 cols), 12..15 -> root.
__global__ __launch_bounds__(32)
void gmm_node_gemm_wmma(const float* __restrict__ x,     // [N,64]
                        const float* __restrict__ g,     // [64,192]
                        const float* __restrict__ root,  // [64,64]
                        const float* __restrict__ bias,  // [64]
                        float* __restrict__ y,           // [N,192] ws
                        float* __restrict__ out,         // [N,64]  gets x@root+bias+x
                        int N)
{
  const int rowTile = blockIdx.x;
  const int colTile = blockIdx.y;          // 0..15
  const int lane    = threadIdx.x;         // 0..31
  const int half    = lane >> 4;           // lane group: K offset 0 or 2
  const int l       = lane & 15;
  const int row0    = rowTile * 16;
  if (row0 + 16 > N) return;               // uniform across wave (EXEC all-1s for WMMA)
  const int  col0   = colTile * 16;
  const bool isRoot = (colTile >= 12);
  const float* B    = isRoot ? root : g;
  const int   ldb   = isRoot ? 64 : YCOLS;
  const int   bcol  = (isRoot ? col0 - YCOLS : col0) + l;

  const float* arow = x + (size_t)(row0 + l) * D_IN;

  v8f c = {};
  #pragma unroll
  for (int k = 0; k < D_IN; k += 4) {
    const int ka = k + half * 2;           // this lane-half's K pair
    v2f a, b;
    // A 16x4: lane l holds row M=row0+l, K = ka, ka+1
    a.x = arow[ka];
    a.y = arow[ka + 1];
    // B 4x16: lane l holds col bcol, K = ka, ka+1
    b.x = B[(size_t)ka * ldb + bcol];
    b.y = B[(size_t)(ka + 1) * ldb + bcol];
    c = __builtin_amdgcn_wmma_f32_16x16x4_f32(
            /*neg_a=*/false, a, /*neg_b=*/false, b,
            /*c_mod=*/(short)0, c, /*reuse_a=*/false, /*reuse_b=*/false);
  }

  // C/D layout: VGPR v -> M = v (lanes 0-15) / v+8 (lanes 16-31), N = l
  if (!isRoot) {
    #pragma unroll
    for (int v = 0; v < 8; ++v) {
      const int r = row0 + v + half * 8;
      y[(size_t)r * YCOLS + col0 + l] = c[v];
    }
  } else {
    const float bv = bias[bcol];
    #pragma unroll
    for (int v = 0; v < 8; ++v) {
      const int r = row0 + v + half * 8;
      out[(size_t)r * D_IN + bcol] = c[v] + bv + x[(size_t)r * D_IN + bcol];
    }
  }
}

// One wave per edge; lane handles feature dims (2*lane, 2*lane+1).
__global__ __launch_bounds__(256)
void gmm_edge_scatter(const float* __restrict__ y,        // [N,192]
                      const long long* __restrict__ srcI, // [E]
                      const long long* __restrict__ dstI, // [E]
                      const float* __restrict__ ea,       // [E]
                      const float* __restrict__ mu,       // [3]
                      const float* __restrict__ sigma,    // [3]
                      float* __restrict__ acc,            // [N,64]
                      float* __restrict__ cnt,            // [N]
                      int E)
{
  const long long tid = (long long)blockIdx.x * blockDim.x + threadIdx.x;
  const int e    = (int)(tid >> 5);
  const int lane = threadIdx.x & 31;
  if (e >= E) return;

  const long long s = srcI[e];
  const long long d = dstI[e];
  const float     p = ea[e];

  float w0, w1, w2;
  { float df = p - mu[0]; float s2 = sigma[0] * sigma[0] + EPS_G;
    w0 = __expf(-0.5f * df * df / s2); }
  { float df = p - mu[1]; float s2 = sigma[1] * sigma[1] + EPS_G;
    w1 = __expf(-0.5f * df * df / s2); }
  { float df = p - mu[2]; float s2 = sigma[2] * sigma[2] + EPS_G;
    w2 = __expf(-0.5f * df * df / s2); }

  const float* yb = y + (size_t)s * YCOLS + lane * 2;   // 8B-aligned -> b64 loads
  const float2 v0 = *(const float2*)(yb);
  const float2 v1 = *(const float2*)(yb + 64);
  const float2 v2 = *(const float2*)(yb + 128);

  const float mx = w0 * v0.x + w1 * v1.x + w2 * v2.x;
  const float my = w0 * v0.y + w1 * v1.y + w2 * v2.y;

  float* ap = acc + (size_t)d * D_IN + lane * 2;
  atomicAdd(ap,     mx);
  atomicAdd(ap + 1, my);
  if (lane == 0) atomicAdd(&cnt[d], 1.0f);
}

// out = silu(acc / max(cnt,1) + out), in place over N*64 elements.
__global__ __launch_bounds__(256)
void gmm_finalize(const float* __restrict__ acc,
                  const float* __restrict__ cnt,
                  float* __restrict__ out, int total)
{
  const int i = blockIdx.x * blockDim.x + threadIdx.x;
  if (i >= total) return;
  const int n = i >> 6;
  const float denom = fmaxf(cnt[n], 1.0f);
  const float z = acc[i] / denom + out[i];
  out[i] = z / (1.0f + __expf(-z));
}

extern "C" void kernel_launch(void* const* d_in, const int* in_sizes, int n_in,
                              void* d_out, int out_size, void* d_ws, size_t ws_size,
                              hipStream_t stream) {
  const float*     x     = (const float*)d_in[0];
  const long long* ei    = (const long long*)d_in[1];   // [2, E] int64
  const float*     ea    = (const float*)d_in[2];
  const float*     g     = (const float*)d_in[3];
  const float*     mu    = (const float*)d_in[4];
  const float*     sigma = (const float*)d_in[5];
  const float*     root  = (const float*)d_in[6];
  const float*     bias  = (const float*)d_in[7];
  float*           out   = (float*)d_out;

  const int N = in_sizes[0] / D_IN;     // 50000
  const int E = in_sizes[1] / 2;        // 800000

  // Workspace layout: y [N*192] | acc [N*64] | cnt [N]  (~51.5 MB)
  float* y   = (float*)d_ws;
  float* acc = y + (size_t)N * YCOLS;
  float* cnt = acc + (size_t)N * D_IN;

  // 1) zero accumulators (acc and cnt are contiguous: N*65 floats)
  hipMemsetAsync(acc, 0, (size_t)N * 65 * sizeof(float), stream);

  // 2) node GEMMs via f32 WMMA: y = x@g, out = x@root + bias + x
  dim3 gGrid((N + 15) / 16, 16);
  gmm_node_gemm_wmma<<<gGrid, dim3(32), 0, stream>>>(x, g, root, bias, y, out, N);

  // 3) per-edge weighted gather + atomic scatter (8 edges per 256-thread block)
  const int eBlocks = (E + 7) / 8;
  gmm_edge_scatter<<<eBlocks, 256, 0, stream>>>(y, ei, ei + E, ea, mu, sigma,
                                                acc, cnt, E);

  // 4) finalize: mean aggregation + residual + SiLU
  const int total = N * D_IN;
  gmm_finalize<<<(total + 255) / 256, 256, 0, stream>>>(acc, cnt, out, total);
}